// GCNClassifier_18056042512835
// MI455X (gfx1250) — compile-verified
//
#include <hip/hip_runtime.h>

typedef __attribute__((ext_vector_type(2))) float v2f;
typedef __attribute__((ext_vector_type(8))) float v8f;

#define NN   50000
#define EE   600000
#define INCH 256
#define HID  128
#define NCLS 2
#define NG   256

// ---------------- degree / normalization ----------------
__global__ void k_init_deg(float* __restrict__ deg) {
  int i = blockIdx.x * blockDim.x + threadIdx.x;
  if (i < NN) deg[i] = 1.0f;  // self-loop
}

__global__ void k_deg_accum(const long long* __restrict__ ei, float* __restrict__ deg) {
  int e = blockIdx.x * blockDim.x + threadIdx.x;
  if (e < EE) {
    int c = (int)ei[(size_t)EE + e];  // col (target)
    atomicAdd(&deg[c], 1.0f);
  }
}

__global__ void k_rsqrt(float* __restrict__ d) {
  int i = blockIdx.x * blockDim.x + threadIdx.x;
  if (i < NN) d[i] = rsqrtf(fmaxf(d[i], 1.0f));
}

// ---------------- B repack into WMMA fragment order ----------------
// Bp element index ((k4*8 + t)*32 + lane) holds the v2f fragment for
// k-step k4, column tile t, lane: { B[k4*4+(lane>>4)*2][t*16+(lane&15)],
//                                   B[k4*4+(lane>>4)*2+1][same col] }
__global__ void k_pack_B(const float* __restrict__ B, v2f* __restrict__ Bp, int K) {
  int idx = blockIdx.x * blockDim.x + threadIdx.x;  // over (K/4)*8*32
  int total = (K >> 2) * 256;
  if (idx >= total) return;
  int lane = idx & 31;
  int t    = (idx >> 5) & 7;
  int k4   = idx >> 8;
  int krow = k4 * 4 + (lane >> 4) * 2;
  int col  = t * 16 + (lane & 15);
  v2f b;
  b.x = B[(size_t)krow * HID + col];
  b.y = B[(size_t)(krow + 1) * HID + col];
  Bp[idx] = b;
}

// ---------------- fp32 WMMA GEMM: C[M,128] = A[M,K] * Bpacked ----------------
// One wave computes a 16-row x 128-col stripe with 8 accumulators.
// B fragments are single coalesced b64 loads; all 8 batched before the WMMAs
// so the scheduler can overlap loads with matrix issue.
__global__ void k_gemm_f32_wmma(const float* __restrict__ A,
                                const v2f* __restrict__ Bp,
                                float* __restrict__ C, int M, int K) {
  const int lane     = threadIdx.x & 31;
  const int wave     = threadIdx.x >> 5;
  const int waveGlob = blockIdx.x * (blockDim.x >> 5) + wave;
  const int row0     = waveGlob * 16;
  if (row0 >= M) return;          // wave-uniform -> EXEC stays all-ones for WMMA
  const int halfSel  = lane >> 4;
  const int lane16   = lane & 15;

  v8f acc[8];
#pragma unroll
  for (int t = 0; t < 8; ++t) acc[t] = (v8f)(0.0f);

  const float* aRow  = A + (size_t)(row0 + lane16) * K + halfSel * 2;
  const v2f*   bBase = Bp + lane;

  for (int k4 = 0; k4 < (K >> 2); ++k4) {
    // A fragment (16x4): one b64 load per lane
    v2f a = *(const v2f*)(aRow + k4 * 4);
    // 8 B fragments: consecutive b64 loads (256B/wave, clause-able)
    v2f bf[8];
#pragma unroll
    for (int t = 0; t < 8; ++t) bf[t] = bBase[(k4 * 8 + t) * 32];
#pragma unroll
    for (int t = 0; t < 8; ++t) {
      acc[t] = __builtin_amdgcn_wmma_f32_16x16x4_f32(
          false, a, false, bf[t], (short)0, acc[t], false, false);
    }
  }

  // C/D layout: VGPR i -> row M=i (lanes 0-15) / M=i+8 (lanes 16-31), N=lane16
#pragma unroll
  for (int t = 0; t < 8; ++t) {
#pragma unroll
    for (int i = 0; i < 8; ++i) {
      int r = row0 + i + halfSel * 8;
      C[(size_t)r * HID + t * 16 + lane16] = acc[t][i];
    }
  }
}

// ---------------- aggregation (self-loop init + edge scatter) ----------------
__global__ void k_agg_selfloop(const float* __restrict__ xw,
                               const float* __restrict__ dinv,
                               float* __restrict__ agg) {
  int idx = blockIdx.x * blockDim.x + threadIdx.x;
  if (idx < NN * HID) {
    int i = idx >> 7;
    float d = dinv[i];
    agg[idx] = xw[idx] * d * d;
  }
}

// one wave per edge; float4 gather + 4 global f32 atomics per lane
__global__ void k_edge_scatter(const long long* __restrict__ ei,
                               const float* __restrict__ xw,
                               const float* __restrict__ dinv,
                               float* __restrict__ agg) {
  int e = blockIdx.x * (blockDim.x >> 5) + (threadIdx.x >> 5);
  if (e >= EE) return;
  int lane = threadIdx.x & 31;
  int r = (int)ei[e];                     // source
  int c = (int)ei[(size_t)EE + e];        // target
  float w = dinv[r] * dinv[c];
  const float4 v = *(const float4*)(xw + (size_t)r * HID + lane * 4);
  float* dst = agg + (size_t)c * HID + lane * 4;
  atomicAdd(dst + 0, v.x * w);
  atomicAdd(dst + 1, v.y * w);
  atomicAdd(dst + 2, v.z * w);
  atomicAdd(dst + 3, v.w * w);
}

__global__ void k_bias_relu(const float* __restrict__ b, float* __restrict__ a) {
  int idx = blockIdx.x * blockDim.x + threadIdx.x;
  if (idx < NN * HID) a[idx] = fmaxf(a[idx] + b[idx & (HID - 1)], 0.0f);
}

// ---------------- global mean pool + linear head ----------------
__global__ void k_pool_zero(float* __restrict__ pool, float* __restrict__ cnt) {
  int idx = blockIdx.x * blockDim.x + threadIdx.x;
  if (idx < NG * HID) pool[idx] = 0.0f;
  if (idx < NG) cnt[idx] = 0.0f;
}

__global__ void k_pool_accum(const float* __restrict__ h,
                             const long long* __restrict__ batch,
                             float* __restrict__ pool, float* __restrict__ cnt) {
  int idx = blockIdx.x * blockDim.x + threadIdx.x;
  if (idx < NN * HID) {
    int i = idx >> 7;
    int c = idx & (HID - 1);
    int g = (int)batch[i];
    atomicAdd(&pool[(size_t)g * HID + c], h[idx]);
    if (c == 0) atomicAdd(&cnt[g], 1.0f);
  }
}

__global__ void k_head(const float* __restrict__ pool, const float* __restrict__ cnt,
                       const float* __restrict__ linW, const float* __restrict__ linb,
                       float* __restrict__ out) {
  int g = blockIdx.x * blockDim.x + threadIdx.x;
  if (g >= NG) return;
  float inv = 1.0f / fmaxf(cnt[g], 1.0f);
  float a0 = linb[0], a1 = linb[1];
  const float* p = pool + (size_t)g * HID;
  for (int c = 0; c < HID; ++c) {
    float v = p[c] * inv;
    a0 += v * linW[c * NCLS + 0];
    a1 += v * linW[c * NCLS + 1];
  }
  out[g * NCLS + 0] = a0;
  out[g * NCLS + 1] = a1;
}

// ---------------- launcher ----------------
extern "C" void kernel_launch(void* const* d_in, const int* in_sizes, int n_in,
                              void* d_out, int out_size, void* d_ws, size_t ws_size,
                              hipStream_t stream) {
  const float*     x     = (const float*)d_in[0];
  const long long* ei    = (const long long*)d_in[1];
  const long long* batch = (const long long*)d_in[2];
  const float*     W1    = (const float*)d_in[3];
  const float*     b1    = (const float*)d_in[4];
  const float*     W2    = (const float*)d_in[5];
  const float*     b2    = (const float*)d_in[6];
  const float*     linW  = (const float*)d_in[7];
  const float*     linb  = (const float*)d_in[8];
  float*           out   = (float*)d_out;

  // workspace layout (256B aligned slices)
  auto align256 = [](size_t v) { return (v + 255) & ~(size_t)255; };
  char* ws = (char*)d_ws;
  size_t o = 0;
  float* dinv  = (float*)(ws + o); o = align256(o + (size_t)NN * 4);
  float* buf1  = (float*)(ws + o); o = align256(o + (size_t)NN * HID * 4);  // xw
  float* buf2  = (float*)(ws + o); o = align256(o + (size_t)NN * HID * 4);  // agg/h
  float* pool  = (float*)(ws + o); o = align256(o + (size_t)NG * HID * 4);
  float* cnt   = (float*)(ws + o); o = align256(o + (size_t)NG * 4);
  v2f*   W1p   = (v2f*)(ws + o);   o = align256(o + (size_t)(INCH / 4) * 256 * 8);
  v2f*   W2p   = (v2f*)(ws + o);   o = align256(o + (size_t)(HID / 4) * 256 * 8);
  (void)ws_size; (void)in_sizes; (void)n_in; (void)out_size;

  const int T = 256;
  const int gN     = (NN + T - 1) / T;
  const int gE     = (EE + T - 1) / T;
  const int gNH    = (NN * HID + T - 1) / T;
  const int gEdgeW = (EE + (T / 32) - 1) / (T / 32);           // 1 wave per edge
  const int wavesM = NN / 16;                                  // 3125
  const int gGemm  = (wavesM + (T / 32) - 1) / (T / 32);       // 8 waves/block

  // 0) repack weights into WMMA fragment order
  k_pack_B<<<((INCH / 4) * 256 + T - 1) / T, T, 0, stream>>>(W1, W1p, INCH);
  k_pack_B<<<((HID / 4) * 256 + T - 1) / T, T, 0, stream>>>(W2, W2p, HID);

  // 1) symmetric normalization dinv = rsqrt(deg)
  k_init_deg<<<gN, T, 0, stream>>>(dinv);
  k_deg_accum<<<gE, T, 0, stream>>>(ei, dinv);
  k_rsqrt<<<gN, T, 0, stream>>>(dinv);

  // 2) layer 1: xw = x @ W1 (fp32 WMMA), aggregate, bias+relu
  k_gemm_f32_wmma<<<gGemm, T, 0, stream>>>(x, W1p, buf1, NN, INCH);
  k_agg_selfloop<<<gNH, T, 0, stream>>>(buf1, dinv, buf2);
  k_edge_scatter<<<gEdgeW, T, 0, stream>>>(ei, buf1, dinv, buf2);
  k_bias_relu<<<gNH, T, 0, stream>>>(b1, buf2);

  // 3) layer 2: xw2 = h1 @ W2, aggregate, bias+relu
  k_gemm_f32_wmma<<<gGemm, T, 0, stream>>>(buf2, W2p, buf1, NN, HID);
  k_agg_selfloop<<<gNH, T, 0, stream>>>(buf1, dinv, buf2);
  k_edge_scatter<<<gEdgeW, T, 0, stream>>>(ei, buf1, dinv, buf2);
  k_bias_relu<<<gNH, T, 0, stream>>>(b2, buf2);

  // 4) global mean pool + linear head
  k_pool_zero<<<(NG * HID + T - 1) / T, T, 0, stream>>>(pool, cnt);
  k_pool_accum<<<gNH, T, 0, stream>>>(buf2, batch, pool, cnt);
  k_head<<<1, NG, 0, stream>>>(pool, cnt, linW, linb, out);
}